// HeteroEnergyGNN_22239340659073
// MI455X (gfx1250) — compile-verified
//
#include <hip/hip_runtime.h>
#include <math.h>

typedef __attribute__((ext_vector_type(2))) float v2f;
typedef __attribute__((ext_vector_type(4))) float v4f;
typedef __attribute__((ext_vector_type(8))) float v8f;

#define D 128
#define GEMM_WAVES 4
#define XPAD 132     // X tile row stride: 16B aligned, 4-bank skew -> conflict free
#define WSTRIDE 288  // packed-W kpair row stride: 256 data + 32 pad -> lanes16-31 hit banks 32-63

// ---------------------------------------------------------------------------
// GEMM: Y[N,128] = X[N,128] @ W[128,128], fp32 WMMA 16x16x4.
// Block = 128 threads (4 waves). W staged in LDS K-pair-packed so each B
// fragment is one aligned ds_load_b64 into a contiguous VGPR pair.
// ---------------------------------------------------------------------------
__global__ void gemm128_wmma(const float* __restrict__ X,
                             const float* __restrict__ W,
                             float* __restrict__ Y, int nRows) {
  extern __shared__ float lds[];
  float* wlds = lds;                          // 64 kpairs * WSTRIDE floats
  float* xlds = lds + 64 * WSTRIDE;           // GEMM_WAVES * 16 * XPAD floats

  const int tid  = threadIdx.x;
  const int lane = tid & 31;
  const int wv   = tid >> 5;

  // cooperative load of W into LDS, packed: wlds[(k>>1)*WSTRIDE + col*2 + (k&1)]
  for (int i = tid; i < D * D / 4; i += blockDim.x) {
    int k  = i >> 5;            // row (K index) 0..127
    int c4 = (i & 31) * 4;      // col base
    v4f v = ((const v4f*)W)[i];
    float* dp = wlds + (size_t)(k >> 1) * WSTRIDE + (k & 1);
    dp[(c4 + 0) * 2] = v.x;
    dp[(c4 + 1) * 2] = v.y;
    dp[(c4 + 2) * 2] = v.z;
    dp[(c4 + 3) * 2] = v.w;
  }

  const int rowBase = (blockIdx.x * GEMM_WAVES + wv) * 16;
  float* xs = xlds + wv * 16 * XPAD;
  const bool active = rowBase < nRows;

  if (active) {
    // stage 16x128 X tile: 512 float4 / 32 lanes = 16 iterations
    for (int it = 0; it < 16; ++it) {
      int idx = it * 32 + lane;
      int r = idx >> 5, c4 = idx & 31;
      v4f v = ((const v4f*)(X + (size_t)(rowBase + r) * D))[c4];
      *(v4f*)(xs + r * XPAD + c4 * 4) = v;
    }
  }
  __syncthreads();
  if (!active) return;

  const int hi = lane >> 4;   // 0: lanes 0-15, 1: lanes 16-31
  const int lo = lane & 15;

  v8f acc[8];
#pragma unroll
  for (int t = 0; t < 8; ++t) acc[t] = (v8f){0.f,0.f,0.f,0.f,0.f,0.f,0.f,0.f};

  const float* xrow = xs + lo * XPAD;
  for (int k = 0; k < D; k += 4) {
    const int ka = k + 2 * hi;
    // A fragment: 16x4, lane holds row M=lo, K pair {ka, ka+1} (aligned b64)
    v2f a = *(const v2f*)(xrow + ka);
    // B fragments: kpair row (ka>>1), contiguous {W[ka][n], W[ka+1][n]} pairs
    const float* wp = wlds + (size_t)((k >> 1) + hi) * WSTRIDE;
#pragma unroll
    for (int t = 0; t < 8; ++t) {
      v2f b = *(const v2f*)(wp + (t * 16 + lo) * 2);
      acc[t] = __builtin_amdgcn_wmma_f32_16x16x4_f32(
          /*neg_a=*/false, a, /*neg_b=*/false, b,
          /*c_mod=*/(short)0, acc[t], /*reuse_a=*/false, /*reuse_b=*/false);
    }
  }

  // D layout: VGPR r -> M = r + 8*hi, N = lo (within 16-col tile t)
#pragma unroll
  for (int t = 0; t < 8; ++t)
#pragma unroll
    for (int r = 0; r < 8; ++r)
      Y[(size_t)(rowBase + r + 8 * hi) * D + t * 16 + lo] = acc[t][r];
}

// ---------------------------------------------------------------------------
// Per-node attention dot: out[n,h] = sum_c H[n, h*32+c] * a[h*32+c], H=4, C=32.
// One wave per node; coalesced 128B loads; butterfly reduction.
// ---------------------------------------------------------------------------
__global__ void rowdot4_kernel(const float* __restrict__ H,
                               const float* __restrict__ a,
                               float* __restrict__ out, int n) {
  int wave = (blockIdx.x * blockDim.x + threadIdx.x) >> 5;
  int lane = threadIdx.x & 31;
  if (wave >= n) return;
  const float* row = H + (size_t)wave * D;
  float p0 = row[lane]        * a[lane];
  float p1 = row[32 + lane]   * a[32 + lane];
  float p2 = row[64 + lane]   * a[64 + lane];
  float p3 = row[96 + lane]   * a[96 + lane];
#pragma unroll
  for (int off = 16; off > 0; off >>= 1) {
    p0 += __shfl_xor(p0, off, 32);
    p1 += __shfl_xor(p1, off, 32);
    p2 += __shfl_xor(p2, off, 32);
    p3 += __shfl_xor(p3, off, 32);
  }
  if (lane == 0) {
    float* o = out + (size_t)wave * 4;
    o[0] = p0; o[1] = p1; o[2] = p2; o[3] = p3;
  }
}

// ---------------------------------------------------------------------------
// float atomic max via sign-split int/uint trick (dest pre-filled with -inf)
// ---------------------------------------------------------------------------
__device__ __forceinline__ void atomicMaxF(float* addr, float v) {
  if (v >= 0.f) atomicMax((int*)addr, __float_as_int(v));
  else          atomicMin((unsigned int*)addr, __float_as_uint(v));
}

__global__ void fillv_kernel(float* p, float v, int n) {
  int i = blockIdx.x * blockDim.x + threadIdx.x;
  if (i < n) p[i] = v;
}

// pass 1: e = leaky_relu(es[src]+ed[dst]); segment max into m[dst]
__global__ void gat_edge_e(const float* __restrict__ es, const float* __restrict__ ed,
                           const int* __restrict__ src, const int* __restrict__ dst,
                           float* __restrict__ e, float* __restrict__ m, int E) {
  int i = blockIdx.x * blockDim.x + threadIdx.x;
  if (i >= E) return;
  int s = src[i], d = dst[i];
#pragma unroll
  for (int h = 0; h < 4; ++h) {
    float v = es[(size_t)s * 4 + h] + ed[(size_t)d * 4 + h];
    v = v > 0.f ? v : 0.2f * v;
    e[(size_t)i * 4 + h] = v;
    atomicMaxF(&m[(size_t)d * 4 + h], v);
  }
}

// pass 2: w = exp(e - m[dst]); segment sum into ssum[dst]
__global__ void gat_edge_w(const float* __restrict__ e, const int* __restrict__ dst,
                           const float* __restrict__ m, float* __restrict__ w,
                           float* __restrict__ ssum, int E) {
  int i = blockIdx.x * blockDim.x + threadIdx.x;
  if (i >= E) return;
  int d = dst[i];
#pragma unroll
  for (int h = 0; h < 4; ++h) {
    float v = __expf(e[(size_t)i * 4 + h] - m[(size_t)d * 4 + h]);
    w[(size_t)i * 4 + h] = v;
    atomicAdd(&ssum[(size_t)d * 4 + h], v);
  }
}

// pass 3: acc[dst, f] += hs[src, f] * alpha   (block = 128 threads = feature dim)
#define EPB 8
__global__ void gat_scatter(const float* __restrict__ hs,
                            const int* __restrict__ src, const int* __restrict__ dst,
                            const float* __restrict__ w, const float* __restrict__ ssum,
                            float* __restrict__ acc, int E) {
  int f = threadIdx.x;      // 0..127
  int h = f >> 5;
  int base = blockIdx.x * EPB;
  int end = base + EPB; if (end > E) end = E;
  for (int i = base; i < end; ++i) {
    int s = src[i], d = dst[i];
    float alpha = w[(size_t)i * 4 + h] / (ssum[(size_t)d * 4 + h] + 1e-16f);
    atomicAdd(&acc[(size_t)d * D + f], hs[(size_t)s * D + f] * alpha);
  }
}

// out = x + scale*(acc + bias[f])
__global__ void combine_res(const float* __restrict__ x, const float* __restrict__ acc,
                            const float* __restrict__ bias, float* __restrict__ out,
                            float scale, size_t n) {
  size_t i = (size_t)blockIdx.x * blockDim.x + threadIdx.x;
  if (i >= n) return;
  int f = (int)(i & (D - 1));
  out[i] = x[i] + scale * (acc[i] + bias[f]);
}

// ---- GCN ----
__global__ void deg_count(const int* __restrict__ dst, float* __restrict__ deg, int E) {
  int i = blockIdx.x * blockDim.x + threadIdx.x;
  if (i < E) atomicAdd(&deg[dst[i]], 1.f);
}

__global__ void make_dinv(const float* __restrict__ deg, float* __restrict__ dinv, int n) {
  int i = blockIdx.x * blockDim.x + threadIdx.x;
  if (i < n) dinv[i] = rsqrtf(deg[i] + 1.f);
}

__global__ void gcn_scatter(const float* __restrict__ xw,
                            const int* __restrict__ src, const int* __restrict__ dst,
                            const float* __restrict__ dinv, float* __restrict__ acc, int E) {
  int f = threadIdx.x;
  int base = blockIdx.x * EPB;
  int end = base + EPB; if (end > E) end = E;
  for (int i = base; i < end; ++i) {
    int s = src[i], d = dst[i];
    float c = dinv[s] * dinv[d];
    atomicAdd(&acc[(size_t)d * D + f], xw[(size_t)s * D + f] * c);
  }
}

// out = x + 0.2*(acc + xw*dinv[n]^2 + bg[f])
__global__ void combine_building(const float* __restrict__ x, const float* __restrict__ acc,
                                 const float* __restrict__ xw, const float* __restrict__ dinv,
                                 const float* __restrict__ bg, float* __restrict__ out,
                                 size_t n) {
  size_t i = (size_t)blockIdx.x * blockDim.x + threadIdx.x;
  if (i >= n) return;
  int f = (int)(i & (D - 1));
  size_t node = i >> 7;
  float di = dinv[node];
  out[i] = x[i] + 0.2f * (acc[i] + xw[i] * di * di + bg[f]);
}

// ---------------------------------------------------------------------------
extern "C" void kernel_launch(void* const* d_in, const int* in_sizes, int n_in,
                              void* d_out, int out_size, void* d_ws, size_t ws_size,
                              hipStream_t stream) {
  const float* x_b  = (const float*)d_in[0];
  const float* x_c  = (const float*)d_in[1];
  const float* x_t  = (const float*)d_in[2];
  const float* Ws1  = (const float*)d_in[3];
  const float* Wd1  = (const float*)d_in[4];
  const float* a_s1 = (const float*)d_in[5];
  const float* a_d1 = (const float*)d_in[6];
  const float* b1   = (const float*)d_in[7];
  const float* Ws2  = (const float*)d_in[8];
  const float* Wd2  = (const float*)d_in[9];
  const float* a_s2 = (const float*)d_in[10];
  const float* a_d2 = (const float*)d_in[11];
  const float* b2   = (const float*)d_in[12];
  const float* Wg   = (const float*)d_in[13];
  const float* bg   = (const float*)d_in[14];
  const int* b2c_src = (const int*)d_in[15];
  const int* b2c_dst = (const int*)d_in[16];
  const int* c2t_src = (const int*)d_in[17];
  const int* c2t_dst = (const int*)d_in[18];
  const int* adj_src = (const int*)d_in[19];
  const int* adj_dst = (const int*)d_in[20];

  const int Nb   = in_sizes[0] / D;
  const int Nc   = in_sizes[1] / D;
  const int Nt   = in_sizes[2] / D;
  const int Ebc  = in_sizes[15];
  const int Ect  = in_sizes[17];
  const int Eadj = in_sizes[19];

  float* out_b = (float*)d_out;                        // [Nb,128]
  float* out_c = out_b + (size_t)Nb * D;               // [Nc,128]
  float* out_t = out_c + (size_t)Nc * D;               // [Nt,128]

  // workspace carve
  float* ws = (float*)d_ws;
  size_t off = 0;
  auto carve = [&](size_t n) { float* p = ws + off; off += n; return p; };
  float* hs1  = carve((size_t)Nb * D);
  float* xwg  = carve((size_t)Nb * D);
  float* accB = carve((size_t)Nb * D);
  float* hd1  = carve((size_t)Nc * D);
  float* hs2  = carve((size_t)Nc * D);
  float* accC = carve((size_t)Nc * D);
  float* hd2  = carve((size_t)Nt * D);
  float* accT = carve((size_t)Nt * D);
  float* es1  = carve((size_t)Nb * 4);
  float* ed1  = carve((size_t)Nc * 4);
  float* es2  = carve((size_t)Nc * 4);
  float* ed2  = carve((size_t)Nt * 4);
  float* e1   = carve((size_t)Ebc * 4);
  float* w1   = carve((size_t)Ebc * 4);
  float* e2   = carve((size_t)Ect * 4);
  float* w2   = carve((size_t)Ect * 4);
  float* m1   = carve((size_t)Nc * 4);
  float* s1   = carve((size_t)Nc * 4);
  float* m2   = carve((size_t)Nt * 4);
  float* s2   = carve((size_t)Nt * 4);
  float* degB = carve((size_t)Nb);
  float* dinv = carve((size_t)Nb);
  (void)ws_size; (void)n_in; (void)out_size;

  const size_t gemmShm =
      (size_t)(64 * WSTRIDE + GEMM_WAVES * 16 * XPAD) * sizeof(float);
  auto cdiv = [](long long a, long long b) { return (int)((a + b - 1) / b); };

  // zero / -inf initialization (every call; capture-safe)
  hipMemsetAsync(accB, 0, (size_t)Nb * D * sizeof(float), stream);
  hipMemsetAsync(accC, 0, (size_t)Nc * D * sizeof(float), stream);
  hipMemsetAsync(accT, 0, (size_t)Nt * D * sizeof(float), stream);
  hipMemsetAsync(s1,   0, (size_t)Nc * 4 * sizeof(float), stream);
  hipMemsetAsync(s2,   0, (size_t)Nt * 4 * sizeof(float), stream);
  hipMemsetAsync(degB, 0, (size_t)Nb * sizeof(float), stream);
  fillv_kernel<<<cdiv(Nc * 4, 256), 256, 0, stream>>>(m1, -INFINITY, Nc * 4);
  fillv_kernel<<<cdiv(Nt * 4, 256), 256, 0, stream>>>(m2, -INFINITY, Nt * 4);

  // ---- stage 1: building -> cable GAT ----
  gemm128_wmma<<<cdiv(Nb, 64), 128, gemmShm, stream>>>(x_b, Ws1, hs1, Nb);
  gemm128_wmma<<<cdiv(Nc, 64), 128, gemmShm, stream>>>(x_c, Wd1, hd1, Nc);
  rowdot4_kernel<<<cdiv(Nb, 8), 256, 0, stream>>>(hs1, a_s1, es1, Nb);
  rowdot4_kernel<<<cdiv(Nc, 8), 256, 0, stream>>>(hd1, a_d1, ed1, Nc);
  gat_edge_e<<<cdiv(Ebc, 256), 256, 0, stream>>>(es1, ed1, b2c_src, b2c_dst, e1, m1, Ebc);
  gat_edge_w<<<cdiv(Ebc, 256), 256, 0, stream>>>(e1, b2c_dst, m1, w1, s1, Ebc);
  gat_scatter<<<cdiv(Ebc, EPB), D, 0, stream>>>(hs1, b2c_src, b2c_dst, w1, s1, accC, Ebc);
  combine_res<<<cdiv((size_t)Nc * D, 256), 256, 0, stream>>>(x_c, accC, b1, out_c, 0.5f,
                                                             (size_t)Nc * D);

  // ---- stage 2: (updated) cable -> transformer GAT ----
  gemm128_wmma<<<cdiv(Nc, 64), 128, gemmShm, stream>>>(out_c, Ws2, hs2, Nc);
  gemm128_wmma<<<cdiv(Nt, 64), 128, gemmShm, stream>>>(x_t, Wd2, hd2, Nt);
  rowdot4_kernel<<<cdiv(Nc, 8), 256, 0, stream>>>(hs2, a_s2, es2, Nc);
  rowdot4_kernel<<<cdiv(Nt, 8), 256, 0, stream>>>(hd2, a_d2, ed2, Nt);
  gat_edge_e<<<cdiv(Ect, 256), 256, 0, stream>>>(es2, ed2, c2t_src, c2t_dst, e2, m2, Ect);
  gat_edge_w<<<cdiv(Ect, 256), 256, 0, stream>>>(e2, c2t_dst, m2, w2, s2, Ect);
  gat_scatter<<<cdiv(Ect, EPB), D, 0, stream>>>(hs2, c2t_src, c2t_dst, w2, s2, accT, Ect);
  combine_res<<<cdiv((size_t)Nt * D, 256), 256, 0, stream>>>(x_t, accT, b2, out_t, 0.5f,
                                                             (size_t)Nt * D);

  // ---- stage 3: building adjacency GCN ----
  gemm128_wmma<<<cdiv(Nb, 64), 128, gemmShm, stream>>>(x_b, Wg, xwg, Nb);
  deg_count<<<cdiv(Eadj, 256), 256, 0, stream>>>(adj_dst, degB, Eadj);
  make_dinv<<<cdiv(Nb, 256), 256, 0, stream>>>(degB, dinv, Nb);
  gcn_scatter<<<cdiv(Eadj, EPB), D, 0, stream>>>(xwg, adj_src, adj_dst, dinv, accB, Eadj);
  combine_building<<<cdiv((size_t)Nb * D, 256), 256, 0, stream>>>(x_b, accB, xwg, dinv, bg,
                                                                  out_b, (size_t)Nb * D);
}